// ResidualBlock_13786845020197
// MI455X (gfx1250) — compile-verified
//
#include <hip/hip_runtime.h>
#include <hip/hip_bf16.h>

#define D 256
#define BN_EPS 1e-5f

typedef __attribute__((ext_vector_type(2))) float v2f;
typedef __attribute__((ext_vector_type(8))) float v8f;

// ---------------- degree / norm ----------------
__global__ void deg_init_kernel(float* __restrict__ deg, int n) {
    int i = blockIdx.x * blockDim.x + threadIdx.x;
    if (i < n) deg[i] = 1.0f;  // self-loop weight 1.0
}

__global__ void deg_accum_kernel(const long long* __restrict__ dst,
                                 const float* __restrict__ w,
                                 float* __restrict__ deg, int e) {
    int i = blockIdx.x * blockDim.x + threadIdx.x;
    if (i < e) atomicAdd(&deg[dst[i]], w[i]);
}

__global__ void dinv_kernel(const float* __restrict__ deg,
                            float* __restrict__ dinv, int n) {
    int i = blockIdx.x * blockDim.x + threadIdx.x;
    if (i < n) {
        float dg = deg[i];
        dinv[i] = dg > 0.0f ? rsqrtf(dg) : 0.0f;
    }
}

// ---------------- GEMM: C[M,256] = A[M,256] @ W[256,256] via V_WMMA_F32_16X16X4_F32
// grid = (M/16, 2), block = 256 (8 waves). Wave w of blockIdx.y handles col tile (y*8+w).
// M must be a multiple of 16 (50000 = 3125*16).
__global__ __launch_bounds__(256) void gemm_wmma_f32_kernel(
    const float* __restrict__ A, const float* __restrict__ W,
    float* __restrict__ C, int M) {
    const int lane = threadIdx.x & 31;
    const int wave = threadIdx.x >> 5;
    const int row0 = blockIdx.x * 16;
    const int col0 = (blockIdx.y * 8 + wave) * 16;

    const int m    = row0 + (lane & 15);     // A-matrix row for this lane
    const int koff = (lane >> 4) * 2;        // ISA 16x4 A layout: lanes16-31 hold K+2/K+3
    const int n    = col0 + (lane & 15);     // B-matrix col for this lane

    const float* __restrict__ arow = A + (size_t)m * D;
    v8f acc = {};
#pragma unroll 8
    for (int k = 0; k < D; k += 4) {
        v2f a, b;
        a.x = arow[k + koff];
        a.y = arow[k + koff + 1];
        b.x = W[(size_t)(k + koff) * D + n];
        b.y = W[(size_t)(k + koff + 1) * D + n];
        acc = __builtin_amdgcn_wmma_f32_16x16x4_f32(
            false, a, false, b, (short)0, acc, false, false);
    }
    // D layout: lane holds col (lane&15), rows (lane>>4)*8 + v
    const int dn  = col0 + (lane & 15);
    const int dm0 = row0 + (lane >> 4) * 8;
#pragma unroll
    for (int v = 0; v < 8; ++v)
        C[(size_t)(dm0 + v) * D + dn] = acc[v];
}

// ---------------- edge aggregation: one wave32 per edge, 8 floats/lane ----------------
__global__ __launch_bounds__(256) void edge_aggregate_kernel(
    const float* __restrict__ feat, const long long* __restrict__ src,
    const long long* __restrict__ dst, const float* __restrict__ w,
    const float* __restrict__ dinv, float* __restrict__ out, int e) {
    const int lane = threadIdx.x & 31;
    const int wave = threadIdx.x >> 5;
    const long long eid = (long long)blockIdx.x * 8 + wave;
    if (eid >= e) return;

    const long long s = src[eid];
    const long long d = dst[eid];
    const float norm = dinv[s] * w[eid] * dinv[d];

    const float4* __restrict__ fs = (const float4*)(feat + (size_t)s * D);
    float* __restrict__ od = out + (size_t)d * D;

    float4 v0 = fs[lane * 2 + 0];
    float4 v1 = fs[lane * 2 + 1];
    const int j = lane * 8;
    atomicAdd(&od[j + 0], v0.x * norm);
    atomicAdd(&od[j + 1], v0.y * norm);
    atomicAdd(&od[j + 2], v0.z * norm);
    atomicAdd(&od[j + 3], v0.w * norm);
    atomicAdd(&od[j + 4], v1.x * norm);
    atomicAdd(&od[j + 5], v1.y * norm);
    atomicAdd(&od[j + 6], v1.z * norm);
    atomicAdd(&od[j + 7], v1.w * norm);
}

// ---------------- layer-1 epilogue: h = relu(agg + xw*dinv^2 + b1), write into xw ----
__global__ void fuse1_kernel(float* __restrict__ xw, const float* __restrict__ agg,
                             const float* __restrict__ dinv,
                             const float* __restrict__ b1, size_t total) {
    size_t i = (size_t)blockIdx.x * blockDim.x + threadIdx.x;
    if (i >= total) return;
    size_t row = i >> 8;
    int col = (int)(i & (D - 1));
    float di = dinv[row];
    float v = agg[i] + xw[i] * di * di + b1[col];
    xw[i] = fmaxf(v, 0.0f);
}

// ---------------- layer-2 self-loop: out2 += xw2 * dinv^2 (b2 cancels in BN) --------
__global__ void selfloop2_kernel(float* __restrict__ out2,
                                 const float* __restrict__ xw2,
                                 const float* __restrict__ dinv, size_t total) {
    size_t i = (size_t)blockIdx.x * blockDim.x + threadIdx.x;
    if (i >= total) return;
    size_t row = i >> 8;
    float di = dinv[row];
    out2[i] += xw2[i] * di * di;
}

// ---------------- BN column stats: per-block partials -> 256-address atomics --------
__global__ __launch_bounds__(256) void bn_stats_kernel(
    const float* __restrict__ out2, float* __restrict__ csum,
    float* __restrict__ csq, int nrows) {
    const int j = threadIdx.x;  // column
    float s = 0.0f, q = 0.0f;
    for (int r = blockIdx.x; r < nrows; r += gridDim.x) {
        float v = out2[(size_t)r * D + j];
        s += v;
        q += v * v;
    }
    atomicAdd(&csum[j], s);
    atomicAdd(&csq[j], q);
}

// ---------------- final: BN (batch stats, biased var) + residual + ReLU, in place ---
__global__ void final_kernel(float* __restrict__ out2, const float* __restrict__ x,
                             const float* __restrict__ gamma,
                             const float* __restrict__ beta,
                             const float* __restrict__ csum,
                             const float* __restrict__ csq,
                             float inv_n, size_t total) {
    size_t i = (size_t)blockIdx.x * blockDim.x + threadIdx.x;
    if (i >= total) return;
    int col = (int)(i & (D - 1));
    float mean = csum[col] * inv_n;
    float var = csq[col] * inv_n - mean * mean;
    float inv = rsqrtf(var + BN_EPS);
    float v = (out2[i] - mean) * inv * gamma[col] + beta[col] + x[i];
    out2[i] = fmaxf(v, 0.0f);
}

extern "C" void kernel_launch(void* const* d_in, const int* in_sizes, int n_in,
                              void* d_out, int out_size, void* d_ws, size_t ws_size,
                              hipStream_t stream) {
    const float* x          = (const float*)d_in[0];
    const long long* ei     = (const long long*)d_in[1];
    const float* ew         = (const float*)d_in[2];
    const float* W1         = (const float*)d_in[3];
    const float* b1         = (const float*)d_in[4];
    const float* W2         = (const float*)d_in[5];
    const float* b2         = (const float*)d_in[6];  // cancels in BN; unused
    const float* gamma      = (const float*)d_in[7];
    const float* beta       = (const float*)d_in[8];
    (void)b2; (void)n_in; (void)ws_size;

    const int n = in_sizes[0] / D;          // 50000
    const int e = in_sizes[2];              // 1600000
    const size_t nd = (size_t)n * D;
    const long long* src = ei;
    const long long* dst = ei + e;

    float* ws   = (float*)d_ws;
    float* xw   = ws;                // [N,D]  layer-1 xw, then h after fuse1
    float* agg  = ws + nd;           // [N,D]  layer-1 agg, then layer-2 xw
    float* deg  = ws + 2 * nd;       // [N]
    float* dinv = deg + n;           // [N]
    float* csum = dinv + n;          // [D]
    float* csq  = csum + D;          // [D]
    float* out2 = (float*)d_out;     // [N,D]  layer-2 accumulator, then final output

    // zero accumulators (graph-capturable async memsets)
    hipMemsetAsync(agg, 0, nd * sizeof(float), stream);
    hipMemsetAsync(out2, 0, nd * sizeof(float), stream);
    hipMemsetAsync(csum, 0, 2 * D * sizeof(float), stream);

    const int tpb = 256;
    const int nblk  = (n + tpb - 1) / tpb;
    const int eblk  = (e + tpb - 1) / tpb;
    const int ndblk = (int)((nd + tpb - 1) / tpb);
    const int ewav  = (e + 7) / 8;  // one wave32 per edge, 8 waves per block

    // degrees (self-loops via init to 1.0) and normalization
    deg_init_kernel<<<nblk, tpb, 0, stream>>>(deg, n);
    deg_accum_kernel<<<eblk, tpb, 0, stream>>>(dst, ew, deg, e);
    dinv_kernel<<<nblk, tpb, 0, stream>>>(deg, dinv, n);

    // layer 1: xw = x @ W1 ; agg = scatter(xw) ; h = relu(agg + selfloop + b1)
    gemm_wmma_f32_kernel<<<dim3(n / 16, 2), tpb, 0, stream>>>(x, W1, xw, n);
    edge_aggregate_kernel<<<ewav, tpb, 0, stream>>>(xw, src, dst, ew, dinv, agg, e);
    fuse1_kernel<<<ndblk, tpb, 0, stream>>>(xw, agg, dinv, b1, nd);

    // layer 2: xw2 = h @ W2 (into agg) ; out2 = scatter(xw2) + selfloop
    gemm_wmma_f32_kernel<<<dim3(n / 16, 2), tpb, 0, stream>>>(xw, W2, agg, n);
    edge_aggregate_kernel<<<ewav, tpb, 0, stream>>>(agg, src, dst, ew, dinv, out2, e);
    selfloop2_kernel<<<ndblk, tpb, 0, stream>>>(out2, agg, dinv, nd);

    // batch-norm stats + fused BN/residual/ReLU (b2 shift cancels exactly in BN)
    bn_stats_kernel<<<512, tpb, 0, stream>>>(out2, csum, csq, n);
    final_kernel<<<ndblk, tpb, 0, stream>>>(out2, x, gamma, beta, csum, csq,
                                            1.0f / (float)n, nd);
}